// OverISS_T_13675175871173
// MI455X (gfx1250) — compile-verified
//
#include <hip/hip_runtime.h>

// ---------------- problem constants ----------------
constexpr int B_ = 2, C_ = 6, F_ = 256, N_ = 2000;
constexpr int T_ = 5, DLY = 1, NITER = 3;
constexpr int NSRC = 4, NBAK = 2;          // n_src = 4, n_chan - n_src = 2
constexpr float EPSc = 1e-3f, MODEL_EPSc = 1e-5f, PB_EPSc = 1e-6f;

typedef __attribute__((ext_vector_type(8)))  float        v8f;
typedef __attribute__((ext_vector_type(2)))  float        v2f;
typedef __attribute__((ext_vector_type(4)))  unsigned int v4u;
typedef __attribute__((ext_vector_type(8)))  int          v8i;
typedef __attribute__((ext_vector_type(4)))  int          v4i;

// ---------------- complex helpers on float2 ----------------
__device__ __forceinline__ float2 cmk(float x, float y){ float2 r; r.x = x; r.y = y; return r; }
__device__ __forceinline__ float2 cadd(float2 a, float2 b){ return cmk(a.x+b.x, a.y+b.y); }
__device__ __forceinline__ float2 csub(float2 a, float2 b){ return cmk(a.x-b.x, a.y-b.y); }
__device__ __forceinline__ float2 cmul(float2 a, float2 b){ return cmk(a.x*b.x - a.y*b.y, a.x*b.y + a.y*b.x); }
__device__ __forceinline__ float  cabs2(float2 a){ return a.x*a.x + a.y*a.y; }
__device__ __forceinline__ float2 cscale(float2 a, float s){ return cmk(a.x*s, a.y*s); }
__device__ __forceinline__ float2 cinv(float2 a){
  float d = 1.0f / fmaxf(cabs2(a), 1e-30f); return cmk(a.x*d, -a.y*d);
}

__device__ __forceinline__ size_t xidx(int b, int c, int f, int n){
  return ((size_t)((b*C_ + c)*F_ + f))*N_ + (size_t)n;
}

// 4x4 complex Gauss-Jordan with partial pivoting, up to 2 RHS, in place.
__device__ void csolve4(float2 M[4][4], float2 R[4][2], int nrhs){
  for (int k = 0; k < 4; ++k){
    int p = k; float best = cabs2(M[k][k]);
    for (int i = k+1; i < 4; ++i){ float m = cabs2(M[i][k]); if (m > best){ best = m; p = i; } }
    if (p != k){
      for (int j = 0; j < 4; ++j){ float2 t = M[k][j]; M[k][j] = M[p][j]; M[p][j] = t; }
      for (int r = 0; r < nrhs; ++r){ float2 t = R[k][r]; R[k][r] = R[p][r]; R[p][r] = t; }
    }
    float2 iv = cinv(M[k][k]);
    for (int j = 0; j < 4; ++j) M[k][j] = cmul(M[k][j], iv);
    for (int r = 0; r < nrhs; ++r) R[k][r] = cmul(R[k][r], iv);
    for (int i = 0; i < 4; ++i) if (i != k){
      float2 fk = M[i][k];
      for (int j = 0; j < 4; ++j) M[i][j] = csub(M[i][j], cmul(fk, M[k][j]));
      for (int r = 0; r < nrhs; ++r) R[i][r] = csub(R[i][r], cmul(fk, R[k][r]));
    }
  }
}

// =====================================================================
// Tensor Data Mover: minimal D# + tensor_load_to_lds issue.
// Loads a (tile0 x tile1) tile of dsz-coded elements (2 -> 4B, 3 -> 8B)
// from a row-major global tensor (dim0 stride stride0 elements) into
// contiguous LDS at ldsAddr. Tracked by TENSORcnt.
// =====================================================================
__device__ __forceinline__ void tdm_load(unsigned ldsAddr, const void* gaddr,
                                         int dsz, int tile0, int tile1,
                                         int tdim0, int tdim1, int stride0){
  unsigned long long ga = (unsigned long long)gaddr;
  v4u g0;
  g0[0] = 1u;                                    // count=1 (valid), user mode
  g0[1] = ldsAddr;                               // lds_addr [63:32]
  g0[2] = (unsigned)ga;                          // global_addr [95:64]
  g0[3] = (unsigned)((ga >> 32) & 0x01ffffffu)   // global_addr [120:96]
        | (2u << 30);                            // type=2 ("image") [127:126]
  v8i g1;
  g1[0] = dsz << 16;                                                   // data_size [17:16]
  g1[1] = (int)((unsigned)(tdim0 & 0xffff) << 16);                     // tensor_dim0 lo16
  g1[2] = (int)(((unsigned)tdim0 >> 16) |
                ((unsigned)(tdim1 & 0xffff) << 16));                   // dim0 hi | dim1 lo
  g1[3] = (int)(((unsigned)tdim1 >> 16) |
                ((unsigned)(tile0 & 0xffff) << 16));                   // dim1 hi | tile_dim0
  g1[4] = (int)(tile1 & 0xffff);                                       // tile_dim1 (tile_dim2=0)
  g1[5] = stride0;                                                     // tensor_dim0_stride lo32
  g1[6] = 0;
  g1[7] = 0;
  v4i g2 = (v4i)0, g3 = (v4i)0;
#if defined(__clang_major__) && (__clang_major__ >= 23)
  v8i g4 = (v8i)0;
  __builtin_amdgcn_tensor_load_to_lds(g0, g1, g2, g3, g4, 0);
#else
  __builtin_amdgcn_tensor_load_to_lds(g0, g1, g2, g3, 0);
#endif
}

// =====================================================================
// Kernel 1: C_XX = X X^H / N via V_WMMA_F32_16X16X4_F32 (one wave/(b,f)).
// Interleaved rows: row 2c = Re(ch c), 2c+1 = Im(ch c). For the 32-bit
// layouts the A-tile and B(=A^T)-tile per-lane data coincide: lane(hl,m)
// holds { A[m][k0+2hl], A[m][k0+2hl+1] } in both roles -> one aligned
// float2 load per lane per WMMA, branch-free (pad rows alias row 0,
// their garbage lands in unread G entries). 2000 % 4 == 0 -> no tail.
// =====================================================================
__global__ void __launch_bounds__(32) k_gram(const float* __restrict__ Xr,
                                             const float* __restrict__ Xi,
                                             float* __restrict__ Cxx){
  const int bf = blockIdx.x, b = bf / F_, f = bf % F_;
  const int lane = threadIdx.x, hl = lane >> 4, mrow = lane & 15;
  const int ch = mrow >> 1, isim = mrow & 1;
  const float* rowp = (ch < C_) ? ((isim ? Xi : Xr) + xidx(b, ch, f, 0))
                                : (Xr + xidx(b, 0, f, 0));   // pad rows alias row 0
  const float* lanep = rowp + 2*hl;

  v8f acc = {};
  #pragma unroll 4
  for (int k0 = 0; k0 < N_; k0 += 4){
    if ((k0 & 255) == 0 && k0 + 320 < N_) __builtin_prefetch(lanep + k0 + 256, 0, 1);
    float2 d = *(const float2*)(lanep + k0);
    v2f a; a[0] = d.x; a[1] = d.y;
    acc = __builtin_amdgcn_wmma_f32_16x16x4_f32(false, a, false, a, (short)0, acc, false, false);
  }

  __shared__ float G[16][16];
  #pragma unroll
  for (int r = 0; r < 8; ++r) G[r + hl*8][mrow] = acc[r];
  __syncthreads();
  const float inv = 1.0f / (float)N_;
  for (int e = lane; e < C_*C_; e += 32){
    int cc = e / C_, dd = e % C_;
    float re = (G[2*cc][2*dd]   + G[2*cc+1][2*dd+1]) * inv;
    float im = (G[2*cc+1][2*dd] - G[2*cc][2*dd+1])   * inv;
    float* o = Cxx + ((size_t)bf*C_*C_ + e)*2;
    o[0] = re; o[1] = im;
  }
}

// =====================================================================
// Kernel 2: C_XbarX[(d,t)][c] = sum_n Xbar[d,t,n] conj(X[c,n]) / N, with
// Xbar[d,t,n] = X[d, n + t - (T+DLY)] (never materialized). 60 real rows
// (re/im interleaved) tiled 4x16; one wave per (b,f,mtile).
// f32 WMMA; only k0 in {0,4} can hit negative shifted indices -> two
// guarded steps, the remaining 498 are branch-free. No tail (2000%4==0,
// and shifted index <= 1999 + tsh <= 1997).
// =====================================================================
template<bool GUARD>
__device__ __forceinline__ v8f cxb_step32(const float* __restrict__ rowp,
                                          const float* __restrict__ colp,
                                          int k0, int tsh, int hl, v8f acc){
  const int base = k0 + 2*hl;
  float a0, a1;
  if (GUARD){
    int n0 = base + tsh;
    a0 = (n0     >= 0) ? rowp[n0]     : 0.0f;
    a1 = (n0 + 1 >= 0) ? rowp[n0 + 1] : 0.0f;
  } else {
    a0 = rowp[base + tsh];
    a1 = rowp[base + tsh + 1];
  }
  float2 bd = *(const float2*)(colp + base);     // aligned
  v2f av; av[0] = a0;   av[1] = a1;
  v2f bv; bv[0] = bd.x; bv[1] = bd.y;
  return __builtin_amdgcn_wmma_f32_16x16x4_f32(false, av, false, bv, (short)0, acc, false, false);
}

__global__ void __launch_bounds__(32) k_cxbarx(const float* __restrict__ Xr,
                                               const float* __restrict__ Xi,
                                               float* __restrict__ Cxb){
  const int gid = blockIdx.x, mt = gid & 3, bf = gid >> 2;
  const int b = bf / F_, f = bf % F_;
  const int lane = threadIdx.x, hl = lane >> 4, mrow = lane & 15;

  const int R = mt*16 + mrow, e = R >> 1, isim = R & 1;
  int tsh = -(T_ + DLY);
  const float* rowp;
  if (e < C_*T_){
    int cch = e / T_, t = e % T_;
    tsh = t - (T_ + DLY);
    rowp = (isim ? Xi : Xr) + xidx(b, cch, f, 0);
  } else {
    rowp = Xr + xidx(b, 0, f, 0);            // pad rows alias ch 0, tap 0
  }
  const int nch = mrow >> 1, nim = mrow & 1;
  const float* colp = (nch < C_) ? ((nim ? Xi : Xr) + xidx(b, nch, f, 0))
                                 : (Xr + xidx(b, 0, f, 0));  // pad cols alias ch 0

  v8f acc = {};
  acc = cxb_step32<true>(rowp, colp, 0, tsh, hl, acc);
  acc = cxb_step32<true>(rowp, colp, 4, tsh, hl, acc);
  #pragma unroll 4
  for (int k0 = 8; k0 < N_; k0 += 4){
    if ((k0 & 255) == 0 && k0 + 320 < N_) __builtin_prefetch(rowp + k0 + 256, 0, 1);
    acc = cxb_step32<false>(rowp, colp, k0, tsh, hl, acc);
  }

  __shared__ float G[16][16];
  #pragma unroll
  for (int r = 0; r < 8; ++r) G[r + hl*8][mrow] = acc[r];
  __syncthreads();
  const float inv = 1.0f / (float)N_;
  for (int idx = lane; idx < 8*C_; idx += 32){
    int i = idx / C_, dd = idx % C_;
    int e2 = mt*8 + i;
    if (e2 < C_*T_){
      float re = (G[2*i][2*dd]   + G[2*i+1][2*dd+1]) * inv;
      float im = (G[2*i+1][2*dd] - G[2*i][2*dd+1])   * inv;
      float* o = Cxb + (((size_t)bf*(C_*T_) + e2)*C_ + dd)*2;
      o[0] = re; o[1] = im;
    }
  }
}

// ---------------- init: W = eye(4,6), H = 0 ----------------
__global__ void k_init_wh(float* __restrict__ Wg, float* __restrict__ Hg){
  int bf = blockIdx.x*blockDim.x + threadIdx.x;
  if (bf >= B_*F_) return;
  float2* W = (float2*)Wg + (size_t)bf*NSRC*C_;
  float2* H = (float2*)Hg + (size_t)bf*NSRC*C_*T_;
  for (int s = 0; s < NSRC; ++s)
    for (int d = 0; d < C_; ++d) W[s*C_+d] = cmk(s == d ? 1.0f : 0.0f, 0.0f);
  for (int i = 0; i < NSRC*C_*T_; ++i) H[i] = cmk(0.0f, 0.0f);
}

// ---------------- init: Y = X[:, :NSRC] (interleaved complex) ----------------
__global__ void k_copy_y(const float* __restrict__ Xr, const float* __restrict__ Xi,
                         float* __restrict__ Yg){
  size_t i = (size_t)blockIdx.x*blockDim.x + threadIdx.x;
  size_t tot = (size_t)B_*NSRC*F_*N_;
  if (i >= tot) return;
  int n = (int)(i % N_); size_t t = i / N_;
  int f = (int)(t % F_); t /= F_;
  int s = (int)(t % NSRC); int b = (int)(t / NSRC);
  size_t xi = xidx(b, s, f, n);
  Yg[i*2]   = Xr[xi];
  Yg[i*2+1] = Xi[xi];
}

// ---------------- per-iter: zero g accumulators ----------------
__global__ void k_zero_g(float* __restrict__ Greg){
  if (threadIdx.x < B_*NSRC) Greg[threadIdx.x] = 0.0f;
}

// ---------------- per-iter: r[b,f,n] (Laplace) + g partial sums ----------------
__global__ void __launch_bounds__(256) k_stats(const float* __restrict__ Yg,
                                               float* __restrict__ Rw,
                                               float* __restrict__ Greg){
  const int bf = blockIdx.x, b = bf / F_, f = bf % F_;
  const int tid = threadIdx.x;
  float ls[NSRC] = {0.f, 0.f, 0.f, 0.f};
  for (int n = tid; n < N_; n += 256){
    float s = 0.0f;
    #pragma unroll
    for (int c = 0; c < NSRC; ++c){
      size_t yi = (((size_t)(b*NSRC + c)*F_ + f)*N_ + n)*2;
      float yr = Yg[yi], yi2 = Yg[yi+1];
      float m = yr*yr + yi2*yi2;
      ls[c] += m; s += m;
    }
    Rw[(size_t)bf*N_ + n] = 1.0f / fmaxf(2.0f*sqrtf(s), MODEL_EPSc);
  }
  __shared__ float red[8*NSRC];
  #pragma unroll
  for (int c = 0; c < NSRC; ++c){
    float v = ls[c];
    v += __shfl_down(v, 16); v += __shfl_down(v, 8);
    v += __shfl_down(v, 4);  v += __shfl_down(v, 2);
    v += __shfl_down(v, 1);
    if ((tid & 31) == 0) red[(tid >> 5)*NSRC + c] = v;
  }
  __syncthreads();
  if (tid < NSRC){
    float t = 0.0f;
    for (int w = 0; w < 8; ++w) t += red[w*NSRC + tid];
    atomicAdd(&Greg[b*NSRC + tid], t);
  }
}

// ---------------- per-iter: finalize g, 1/sqrt(g) ----------------
__global__ void k_gfin(float* __restrict__ Greg){
  int t = threadIdx.x;
  if (t < B_*NSRC){
    float g = fmaxf(Greg[t] / (float)((size_t)F_*N_), EPSc);
    Greg[8 + t]  = g;           // g (weights multiplier)
    Greg[16 + t] = rsqrtf(g);   // 1/sqrt(g) (Y,W,H scaling)
  }
}

// ---------------- per-iter: scale W,H; background solve -> J ----------------
__global__ void k_bg(float* __restrict__ Wg, float* __restrict__ Hg,
                     const float* __restrict__ Cxx, const float* __restrict__ Cxb,
                     float* __restrict__ Jg, const float* __restrict__ Greg){
  int bf = blockIdx.x*blockDim.x + threadIdx.x;
  if (bf >= B_*F_) return;
  int b = bf / F_;
  const float* gi = Greg + 16;
  float2* W = (float2*)Wg + (size_t)bf*NSRC*C_;
  float2* H = (float2*)Hg + (size_t)bf*NSRC*C_*T_;
  const float2* CX  = (const float2*)Cxx + (size_t)bf*C_*C_;
  const float2* CXB = (const float2*)Cxb + (size_t)bf*C_*T_*C_;

  float2 A[NSRC][C_];
  for (int s = 0; s < NSRC; ++s)
    for (int j = 0; j < C_; ++j) A[s][j] = cmk(0.f, 0.f);

  for (int s = 0; s < NSRC; ++s){
    float sc = gi[b*NSRC + s];
    for (int c = 0; c < C_; ++c){
      float2 w = cscale(W[s*C_+c], sc);
      W[s*C_+c] = w;                                 // write back scaled W
      for (int j = 0; j < C_; ++j) A[s][j] = cadd(A[s][j], cmul(w, CX[c*C_+j]));
    }
    for (int d = 0; d < C_; ++d)
      for (int t = 0; t < T_; ++t){
        float2 h = cscale(H[(s*C_+d)*T_ + t], sc);
        H[(s*C_+d)*T_ + t] = h;                      // write back scaled H
        const float2* row = CXB + (size_t)(d*T_ + t)*C_;
        for (int j = 0; j < C_; ++j) A[s][j] = cadd(A[s][j], cmul(h, row[j]));
      }
  }
  float2 M[4][4], R[4][2];
  float diag = 0.0f;
  for (int s = 0; s < NSRC; ++s){
    for (int d = 0; d < NSRC; ++d) M[s][d] = A[s][d];
    for (int k = 0; k < NBAK; ++k) R[s][k] = A[s][NSRC + k];
    diag += sqrtf(cabs2(A[s][s]));
  }
  float load = fmaxf(diag*EPSc, EPSc);
  for (int s = 0; s < NSRC; ++s) M[s][s].x += load;
  csolve4(M, R, NBAK);
  float2* J = (float2*)Jg + (size_t)bf*NBAK*NSRC;
  for (int r = 0; r < NBAK; ++r)
    for (int s = 0; s < NSRC; ++s) J[r*NSRC + s] = cmk(R[s][r].x, -R[s][r].y);  // conj
}

// ---------------- per-iter: Z = J X[:, :4] - X[:, 4:] (SoA planes) ----------------
__global__ void __launch_bounds__(256) k_z(const float* __restrict__ Xr,
                                           const float* __restrict__ Xi,
                                           const float* __restrict__ Jg,
                                           float* __restrict__ Zr,
                                           float* __restrict__ Zi){
  size_t i = (size_t)blockIdx.x*blockDim.x + threadIdx.x;
  size_t tot = (size_t)B_*NBAK*F_*N_;
  if (i >= tot) return;
  int n = (int)(i % N_); size_t t = i / N_;
  int f = (int)(t % F_); t /= F_;
  int r = (int)(t % NBAK); int b = (int)(t / NBAK);
  int bf = b*F_ + f;
  const float2* J = (const float2*)Jg + (size_t)bf*NBAK*NSRC + r*NSRC;
  float2 acc = cmk(0.f, 0.f);
  #pragma unroll
  for (int s = 0; s < NSRC; ++s){
    size_t xi = xidx(b, s, f, n);
    acc = cadd(acc, cmul(J[s], cmk(Xr[xi], Xi[xi])));
  }
  size_t xt = xidx(b, NSRC + r, f, n);
  acc = csub(acc, cmk(Xr[xt], Xi[xt]));
  Zr[i] = acc.x; Zi[i] = acc.y;
}

// ---------------- ISS core: 9 reductions + rank-1 Y update ----------------
// SoA pivot (re/im planes); wave32 shuffle reduce + 8-way LDS combine.
__device__ __forceinline__ void iss_core(int tid, float2* Ys,
                                         const float* pr, const float* pi,
                                         const float* rw, float* red, float* vres,
                                         const float* gcl, float2* vv,
                                         float norm, int src_special){
  float acc[9];
  #pragma unroll
  for (int k = 0; k < 9; ++k) acc[k] = 0.0f;
  for (int n = tid; n < N_; n += 256){
    float sx = pr[n], sy = pi[n], r = rw[n];
    acc[8] += r*(sx*sx + sy*sy);
    #pragma unroll
    for (int c = 0; c < NSRC; ++c){
      float2 y = Ys[c*N_ + n];
      acc[2*c]   += r*(y.x*sx + y.y*sy);   // Re( y * conj(s) )
      acc[2*c+1] += r*(y.y*sx - y.x*sy);   // Im( y * conj(s) )
    }
  }
  #pragma unroll
  for (int k = 0; k < 9; ++k){
    float v = acc[k];
    v += __shfl_down(v, 16); v += __shfl_down(v, 8);
    v += __shfl_down(v, 4);  v += __shfl_down(v, 2);
    v += __shfl_down(v, 1);
    if ((tid & 31) == 0) red[(tid >> 5)*9 + k] = v;
  }
  __syncthreads();
  if (tid < 9){
    float t = 0.0f;
    #pragma unroll
    for (int w = 0; w < 8; ++w) t += red[w*9 + tid];
    vres[tid] = t;
  }
  __syncthreads();
  if (tid == 0){
    float rss = vres[8]*norm;
    for (int c = 0; c < NSRC; ++c){
      float den = fmaxf(gcl[c]*rss, EPSc);
      vv[c] = cmk(gcl[c]*vres[2*c]*norm/den, gcl[c]*vres[2*c+1]*norm/den);
    }
    if (src_special >= 0){
      float d = fmaxf(gcl[src_special]*rss, EPSc);
      vv[src_special] = cmk(1.0f - rsqrtf(d), 0.0f);
    }
  }
  __syncthreads();
  float2 v0 = vv[0], v1 = vv[1], v2 = vv[2], v3 = vv[3];
  for (int n = tid; n < N_; n += 256){
    float2 s = cmk(pr[n], pi[n]);
    float2 y;
    y = Ys[0*N_+n]; Ys[0*N_+n] = csub(y, cmul(v0, s));
    y = Ys[1*N_+n]; Ys[1*N_+n] = csub(y, cmul(v1, s));
    y = Ys[2*N_+n]; Ys[2*N_+n] = csub(y, cmul(v2, s));
    y = Ys[3*N_+n]; Ys[3*N_+n] = csub(y, cmul(v3, s));
  }
}

// pivot u in [0, NBAK + C_*T_): u < NBAK -> Z row u; else tap e = u-NBAK,
// channel e/T_, tap e%T_, with lead = (T_+DLY) - tap zero head entries.
__device__ __forceinline__ void issue_pivot(int u, int b, int f,
                                            const float* Xr, const float* Xi,
                                            const float* Zr, const float* Zi,
                                            unsigned reOff, unsigned imOff){
  if (u < NBAK){
    size_t o = ((size_t)(b*NBAK + u)*F_ + f)*N_;
    tdm_load(reOff, Zr + o, 2, N_, 1, N_, 1, N_);
    tdm_load(imOff, Zi + o, 2, N_, 1, N_, 1, N_);
  } else {
    int e = u - NBAK, t = e % T_, lead = (T_ + DLY) - t;
    size_t o = xidx(b, e / T_, f, 0);
    tdm_load(reOff + 4u*(unsigned)lead, Xr + o, 2, N_ - lead, 1, N_ - lead, 1, N_ - lead);
    tdm_load(imOff + 4u*(unsigned)lead, Xi + o, 2, N_ - lead, 1, N_ - lead, 1, N_ - lead);
  }
}

// ---------------- per-iter: all 36 sequential ISS updates, LDS-resident ----------------
// TDM stages the Y tile (2D descriptor), the r row, and double-buffered
// pivot rows one update ahead of the compute (TENSORcnt pipelining).
__global__ void __launch_bounds__(256) k_iss(const float* __restrict__ Xr,
                                             const float* __restrict__ Xi,
                                             float* __restrict__ Yg,
                                             float* __restrict__ Wg,
                                             float* __restrict__ Hg,
                                             const float* __restrict__ Zr,
                                             const float* __restrict__ Zi,
                                             const float* __restrict__ Jg,
                                             const float* __restrict__ Rw,
                                             const float* __restrict__ Greg){
  const int bf = blockIdx.x, b = bf / F_, f = bf % F_;
  const int tid = threadIdx.x;

  extern __shared__ float sm[];
  float2* Ys     = (float2*)sm;               // NSRC*N_ float2
  float*  pivRe0 = (float*)(Ys + NSRC*N_);    // N_
  float*  pivIm0 = pivRe0 + N_;               // N_
  float*  pivRe1 = pivIm0 + N_;               // N_
  float*  pivIm1 = pivRe1 + N_;               // N_
  float*  rw     = pivIm1 + N_;               // N_
  float*  red    = rw + N_;                   // 80 (8 waves x 9, padded)
  float2* Wl     = (float2*)(red + 80);       // NSRC*C_
  float2* Hl     = Wl + NSRC*C_;              // NSRC*C_*T_
  float2* Jl     = Hl + NSRC*C_*T_;           // NBAK*NSRC
  float2* vv     = Jl + NBAK*NSRC;            // NSRC
  float2* tmpW   = vv + NSRC;                 // C_
  float2* tmpH   = tmpW + C_;                 // C_*T_
  float*  vres   = (float*)(tmpH + C_*T_);    // 16
  float*  gcl    = vres + 16;                 // 16

  float* pivRe[2] = { pivRe0, pivRe1 };
  float* pivIm[2] = { pivIm0, pivIm1 };
  // LDS byte addresses for TDM destinations (flat low bits == LDS offset)
  const unsigned ysOff = (unsigned)(size_t)(void*)Ys;
  const unsigned rwOff = (unsigned)(size_t)(void*)rw;
  const unsigned reOff[2] = { (unsigned)(size_t)(void*)pivRe0, (unsigned)(size_t)(void*)pivRe1 };
  const unsigned imOff[2] = { (unsigned)(size_t)(void*)pivIm0, (unsigned)(size_t)(void*)pivIm1 };

  const float* gvv = Greg + 8;
  const float* giv = Greg + 16;

  // Stage Y tile (4 rows, global row stride F_*N_ 8B-elements), r row, and
  // pivot 0 via the Tensor Data Mover (wave 0 issues; TDM ignores EXEC).
  if (tid < 32){
    const float2* ybase = (const float2*)Yg + ((size_t)(b*NSRC)*F_ + f)*N_;
    tdm_load(ysOff, ybase, 3, N_, NSRC, N_, NSRC, F_*N_);
    tdm_load(rwOff, Rw + (size_t)bf*N_, 2, N_, 1, N_, 1, N_);
    issue_pivot(0, b, f, Xr, Xi, Zr, Zi, reOff[0], imOff[0]);
    __builtin_amdgcn_s_wait_tensorcnt(2);   // Y + r done; pivot 0 still in flight
  }
  if (tid < NSRC) gcl[tid] = gvv[b*NSRC + tid];
  if (tid < NSRC*C_)     Wl[tid] = ((const float2*)Wg)[(size_t)bf*NSRC*C_ + tid];
  if (tid < NSRC*C_*T_)  Hl[tid] = ((const float2*)Hg)[(size_t)bf*NSRC*C_*T_ + tid];
  if (tid < NBAK*NSRC)   Jl[tid] = ((const float2*)Jg)[(size_t)bf*NBAK*NSRC + tid];
  __syncthreads();
  // scale Y tile in place by 1/sqrt(g)
  for (int c = 0; c < NSRC; ++c){
    float gi = giv[b*NSRC + c];
    for (int n = tid; n < N_; n += 256) Ys[c*N_ + n] = cscale(Ys[c*N_ + n], gi);
  }
  __syncthreads();

  // ---- source updates (norm = 1/N); pivot buffer 1 is scratch here ----
  for (int src = 0; src < NSRC; ++src){
    for (int n = tid; n < N_; n += 256){
      float2 y = Ys[src*N_ + n];
      pivRe1[n] = y.x; pivIm1[n] = y.y;
    }
    __syncthreads();
    iss_core(tid, Ys, pivRe1, pivIm1, rw, red, vres, gcl, vv, 1.0f/(float)N_, src);
    __syncthreads();
    if (tid < C_)     tmpW[tid] = Wl[src*C_ + tid];
    if (tid < C_*T_)  tmpH[tid] = Hl[src*C_*T_ + tid];
    __syncthreads();
    if (tid < NSRC*C_){
      int c = tid / C_, d = tid % C_;
      Wl[tid] = csub(Wl[tid], cmul(vv[c], tmpW[d]));
    }
    if (tid < NSRC*C_*T_){
      int c = tid / (C_*T_), dt = tid % (C_*T_);
      Hl[tid] = csub(Hl[tid], cmul(vv[c], tmpH[dt]));
    }
    __syncthreads();
  }

  // ---- TDM-pipelined pivot updates: 2 background rows + 30 tap rows ----
  const int NUPD = NBAK + C_*T_;
  for (int u = 0; u < NUPD; ++u){
    const int cur = u & 1;
    if (tid < 32){
      __builtin_amdgcn_s_wait_tensorcnt(0);           // pivot u landed
      if (u + 1 < NUPD)
        issue_pivot(u + 1, b, f, Xr, Xi, Zr, Zi, reOff[cur ^ 1], imOff[cur ^ 1]);
    }
    __syncthreads();                                  // publish TDM data
    int lead = (u < NBAK) ? 0 : (T_ + DLY) - ((u - NBAK) % T_);
    if (tid < lead){ pivRe[cur][tid] = 0.0f; pivIm[cur][tid] = 0.0f; }
    __syncthreads();
    iss_core(tid, Ys, pivRe[cur], pivIm[cur], rw, red, vres, gcl, vv, 1.0f, -1);
    __syncthreads();
    if (u < NBAK){
      if (tid < NSRC*NSRC){
        int c = tid / NSRC, d = tid % NSRC;
        Wl[c*C_ + d] = csub(Wl[c*C_ + d], cmul(vv[c], Jl[u*NSRC + d]));
      }
      if (tid < NSRC)
        Wl[tid*C_ + NSRC + u] = cadd(Wl[tid*C_ + NSRC + u], vv[tid]);
    } else {
      int e = u - NBAK, sc = e / T_, t = e % T_;
      if (tid < NSRC)
        Hl[tid*C_*T_ + sc*T_ + t] = cadd(Hl[tid*C_*T_ + sc*T_ + t], vv[tid]);
    }
    __syncthreads();
  }

  // ---- write back ----
  for (int c = 0; c < NSRC; ++c){
    float2* yrow = (float2*)Yg + ((size_t)(b*NSRC + c)*F_ + f)*N_;
    for (int n = tid; n < N_; n += 256) yrow[n] = Ys[c*N_ + n];
  }
  if (tid < NSRC*C_)    ((float2*)Wg)[(size_t)bf*NSRC*C_ + tid] = Wl[tid];
  if (tid < NSRC*C_*T_) ((float2*)Hg)[(size_t)bf*NSRC*C_*T_ + tid] = Hl[tid];
}

// ---------------- projection back: per-(b,f) 4x4 solve -> a ----------------
__global__ void k_pb(const float* __restrict__ Wg, const float* __restrict__ Jg,
                     float* __restrict__ Ag){
  int bf = blockIdx.x*blockDim.x + threadIdx.x;
  if (bf >= B_*F_) return;
  const float2* W = (const float2*)Wg + (size_t)bf*NSRC*C_;
  const float2* J = (const float2*)Jg + (size_t)bf*NBAK*NSRC;
  float2 Bm[NSRC][NSRC];
  for (int s = 0; s < NSRC; ++s)
    for (int k = 0; k < NSRC; ++k){
      float2 a = W[s*C_ + k];                       // W1
      for (int c = 0; c < NBAK; ++c)
        a = cadd(a, cmul(W[s*C_ + NSRC + c], J[c*NSRC + k]));  // W2 J
      Bm[s][k] = a;
    }
  float2 M[4][4], R[4][2];
  for (int kk = 0; kk < NSRC; ++kk)
    for (int s = 0; s < NSRC; ++s) M[kk][s] = Bm[s][kk];       // BT
  for (int i = 0; i < NSRC; ++i) M[i][i].x += PB_EPSc;
  for (int s = 0; s < NSRC; ++s){ R[s][0] = cmk(s == 0 ? 1.f : 0.f, 0.f); R[s][1] = cmk(0.f,0.f); }
  csolve4(M, R, 1);
  float2* A = (float2*)Ag + (size_t)bf*NSRC;
  for (int s = 0; s < NSRC; ++s) A[s] = R[s][0];
}

// ---------------- out = Y * a (interleaved complex64) ----------------
__global__ void __launch_bounds__(256) k_out(const float* __restrict__ Yg,
                                             const float* __restrict__ Ag,
                                             float* __restrict__ out){
  size_t i = (size_t)blockIdx.x*blockDim.x + threadIdx.x;
  size_t tot = (size_t)B_*NSRC*F_*N_;
  if (i >= tot) return;
  size_t t = i / N_;
  int f = (int)(t % F_); t /= F_;
  int s = (int)(t % NSRC); int b = (int)(t / NSRC);
  const float2* A = (const float2*)Ag + ((size_t)(b*F_ + f))*NSRC + s;
  float2 y = ((const float2*)Yg)[i];
  float2 r = cmul(y, *A);
  out[i*2] = r.x; out[i*2+1] = r.y;
}

// =====================================================================
// Host orchestration
// =====================================================================
extern "C" void kernel_launch(void* const* d_in, const int* in_sizes, int n_in,
                              void* d_out, int out_size, void* d_ws, size_t ws_size,
                              hipStream_t stream){
  (void)in_sizes; (void)n_in; (void)out_size; (void)ws_size;
  const float* Xr = (const float*)d_in[0];
  const float* Xi = (const float*)d_in[1];
  float* out = (float*)d_out;
  float* ws  = (float*)d_ws;

  // workspace layout (floats)
  constexpr size_t SZ_Y  = (size_t)B_*NSRC*F_*N_*2;
  constexpr size_t SZ_ZP = (size_t)B_*NBAK*F_*N_;     // one Z plane
  constexpr size_t SZ_R  = (size_t)B_*F_*N_;
  constexpr size_t SZ_W  = (size_t)B_*F_*NSRC*C_*2;
  constexpr size_t SZ_H  = (size_t)B_*F_*NSRC*C_*T_*2;
  constexpr size_t SZ_CXX= (size_t)B_*F_*C_*C_*2;
  constexpr size_t SZ_CXB= (size_t)B_*F_*C_*T_*C_*2;
  constexpr size_t SZ_J  = (size_t)B_*F_*NBAK*NSRC*2;
  constexpr size_t SZ_A  = (size_t)B_*F_*NSRC*2;

  float* Y    = ws;                  // evolving demixed signal (complex interleaved)
  float* Zr   = Y    + SZ_Y;         // Z real plane
  float* Zi   = Zr   + SZ_ZP;        // Z imag plane
  float* Rw   = Zi   + SZ_ZP;        // Laplace weights r(b,f,n)
  float* W    = Rw   + SZ_R;
  float* H    = W    + SZ_W;
  float* Cxx  = H    + SZ_H;
  float* Cxb  = Cxx  + SZ_CXX;
  float* J    = Cxb  + SZ_CXB;
  float* Asc  = J    + SZ_J;
  float* Greg = Asc  + SZ_A;         // [0:8) accum, [8:16) g, [16:24) 1/sqrt(g)

  const size_t smemIss =
      (size_t)(NSRC*N_)*sizeof(float2) +                       // Ys
      (size_t)(5*N_ + 80)*sizeof(float) +                      // 2x SoA pivots, rw, red
      (size_t)(NSRC*C_ + NSRC*C_*T_ + NBAK*NSRC + NSRC + C_ + C_*T_)*sizeof(float2) +
      (size_t)32*sizeof(float);
  (void)hipFuncSetAttribute((const void*)k_iss,
                            hipFuncAttributeMaxDynamicSharedMemorySize, (int)smemIss);

  const int BF = B_*F_;
  const size_t NEY = (size_t)B_*NSRC*F_*N_;
  const size_t NEZ = (size_t)B_*NBAK*F_*N_;

  k_init_wh<<<(BF + 255)/256, 256, 0, stream>>>(W, H);
  k_copy_y<<<(unsigned)((NEY + 255)/256), 256, 0, stream>>>(Xr, Xi, Y);
  k_gram  <<<BF,   32, 0, stream>>>(Xr, Xi, Cxx);
  k_cxbarx<<<BF*4, 32, 0, stream>>>(Xr, Xi, Cxb);

  for (int it = 0; it < NITER; ++it){
    k_zero_g<<<1, 32, 0, stream>>>(Greg);
    k_stats<<<BF, 256, 0, stream>>>(Y, Rw, Greg);
    k_gfin<<<1, 32, 0, stream>>>(Greg);
    k_bg<<<(BF + 255)/256, 256, 0, stream>>>(W, H, Cxx, Cxb, J, Greg);
    k_z<<<(unsigned)((NEZ + 255)/256), 256, 0, stream>>>(Xr, Xi, J, Zr, Zi);
    k_iss<<<BF, 256, smemIss, stream>>>(Xr, Xi, Y, W, H, Zr, Zi, J, Rw, Greg);
  }

  k_pb<<<(BF + 255)/256, 256, 0, stream>>>(W, J, Asc);
  k_out<<<(unsigned)((NEY + 255)/256), 256, 0, stream>>>(Y, Asc, out);
}